// EgoMotionHead_37546604101724
// MI455X (gfx1250) — compile-verified
//
#include <hip/hip_runtime.h>
#include <hip/hip_bf16.h>
#include <math.h>

typedef __attribute__((ext_vector_type(2))) float v2f;
typedef __attribute__((ext_vector_type(8))) float v8f;

#define TILE 128          // points per wave per iteration
#define WAVES 8           // 256 threads / block
#define ROWSTRIDE 132     // 128 + 4 pad (keeps 16B row alignment: 132*4=528=33*16)

// ---------------------------------------------------------------------------
// Kernel 0: zero the per-batch 4x4 moment accumulators in workspace
// ---------------------------------------------------------------------------
__global__ void zero_ws_kernel(float* mom, int n) {
    int i = blockIdx.x * blockDim.x + threadIdx.x;
    if (i < n) mom[i] = 0.0f;
}

// ---------------------------------------------------------------------------
// Kernel 1: streaming reduction of 4x4 weighted moment matrix per batch.
//   p = (w*xs0, w*xs1, w*xs2, w),  q = (xt0, xt1, xt2, 1),  w = weights*|label-1|
//   Moment = sum_n p(n) q(n)^T   computed via V_WMMA_F32_16X16X4_F32 with
//   4 point-groups packed along the 16x16 diagonal (16 points per WMMA).
// ---------------------------------------------------------------------------
__global__ void moments_kernel(const float* __restrict__ xyz_s,
                               const float* __restrict__ weighted_t,
                               const float* __restrict__ weights,
                               const float* __restrict__ label,
                               float* __restrict__ mom,
                               int N) {
    __shared__ float lds[WAVES][2][4][ROWSTRIDE]; // [wave][P/Q][component][point]
    __shared__ float blockAcc[16];

    const int tid  = threadIdx.x;
    const int wave = tid >> 5;
    const int lane = tid & 31;
    const int b    = blockIdx.y;

    if (tid < 16) blockAcc[tid] = 0.0f;
    __syncthreads();

    const float* xs = xyz_s      + (size_t)b * 3 * N;
    const float* xt = weighted_t + (size_t)b * 3 * N;
    const float* wg = weights    + (size_t)b * N;
    const float* lb = label      + (size_t)b * N;

    float (*P)[ROWSTRIDE] = lds[wave][0];
    float (*Q)[ROWSTRIDE] = lds[wave][1];

    v8f acc = {0.0f, 0.0f, 0.0f, 0.0f, 0.0f, 0.0f, 0.0f, 0.0f};

    const int ntiles      = (N + TILE - 1) / TILE;
    const int nWavesTotal = gridDim.x * WAVES;
    const int waveGlobal  = blockIdx.x * WAVES + wave;

    // Fragment addressing (constant per lane), per ISA 16x4 f32 A layout:
    // lanes 0-15 hold K=0,1 ; lanes 16-31 hold K=2,3.
    const int l15  = lane & 15;
    const int half = lane >> 4;
    const int comp = l15 & 3;                 // p/q component (row/col in 4x4 block)
    const int grp  = l15 >> 2;                // diagonal group 0..3
    const int fbase = 4 * grp + 2 * half;     // first of 2 consecutive points this lane reads

    for (int tile = waveGlobal; tile < ntiles; tile += nWavesTotal) {
        const int n0 = tile * TILE + lane * 4;

        // Wave-uniform prefetch decision (no per-lane exec-mask churn).
        if (tile + nWavesTotal < ntiles) {
            const int npf = n0 + nWavesTotal * TILE;
            __builtin_prefetch(xs + npf, 0, 1);
            __builtin_prefetch(xt + npf, 0, 1);
        }

        float4 x0, x1, x2, t0, t1, t2, wv, lv;
        // Single wave-uniform bounds test for all 8 planes: fast path is
        // 8 straight global_load_b128 with full EXEC, no divergence.
        if (tile * TILE + TILE <= N) {
            x0 = *(const float4*)(xs + 0 * N + n0);
            x1 = *(const float4*)(xs + 1 * N + n0);
            x2 = *(const float4*)(xs + 2 * N + n0);
            t0 = *(const float4*)(xt + 0 * N + n0);
            t1 = *(const float4*)(xt + 1 * N + n0);
            t2 = *(const float4*)(xt + 2 * N + n0);
            wv = *(const float4*)(wg + n0);
            lv = *(const float4*)(lb + n0);
        } else {
            // Tail tile (runs at most once per wave): per-element guards.
            float4 z; z.x = z.y = z.z = z.w = 0.0f;
            x0 = x1 = x2 = t0 = t1 = t2 = wv = lv = z;
            for (int s = 0; s < 4; ++s) {
                const int n = n0 + s;
                if (n < N) {
                    (&x0.x)[s] = xs[0 * N + n];
                    (&x1.x)[s] = xs[1 * N + n];
                    (&x2.x)[s] = xs[2 * N + n];
                    (&t0.x)[s] = xt[0 * N + n];
                    (&t1.x)[s] = xt[1 * N + n];
                    (&t2.x)[s] = xt[2 * N + n];
                    (&wv.x)[s] = wg[n];
                    (&lv.x)[s] = lb[n];
                }
            }
        }

        float4 w;
        w.x = wv.x * fabsf(lv.x - 1.0f);
        w.y = wv.y * fabsf(lv.y - 1.0f);
        w.z = wv.z * fabsf(lv.z - 1.0f);
        w.w = wv.w * fabsf(lv.w - 1.0f);

        // Stage p/q to LDS, component-major. (Out-of-range points have w=0.)
        const int pcol = lane * 4;
        float4 v;
        v.x = w.x * x0.x; v.y = w.y * x0.y; v.z = w.z * x0.z; v.w = w.w * x0.w;
        *(float4*)&P[0][pcol] = v;
        v.x = w.x * x1.x; v.y = w.y * x1.y; v.z = w.z * x1.z; v.w = w.w * x1.w;
        *(float4*)&P[1][pcol] = v;
        v.x = w.x * x2.x; v.y = w.y * x2.y; v.z = w.z * x2.z; v.w = w.w * x2.w;
        *(float4*)&P[2][pcol] = v;
        *(float4*)&P[3][pcol] = w;
        *(float4*)&Q[0][pcol] = t0;
        *(float4*)&Q[1][pcol] = t1;
        *(float4*)&Q[2][pcol] = t2;
        v.x = 1.0f; v.y = 1.0f; v.z = 1.0f; v.w = 1.0f;
        *(float4*)&Q[3][pcol] = v;

        // 8 WMMAs cover the 128-point tile: each contracts K=4 over 4 diagonal
        // groups. Intra-wave LDS RAW ordering handled by DScnt (no barrier).
#pragma unroll
        for (int wb = 0; wb < 8; ++wb) {
            const int pt = wb * 16 + fbase;
            v2f afrag = *(const v2f*)&P[comp][pt];
            v2f bfrag = *(const v2f*)&Q[comp][pt];
            acc = __builtin_amdgcn_wmma_f32_16x16x4_f32(
                /*neg_a=*/false, afrag, /*neg_b=*/false, bfrag,
                /*c_mod=*/(short)0, acc, /*reuse_a=*/false, /*reuse_b=*/false);
        }
    }

    // Extract useful diagonal-block entries.
    // C layout: lanes 0-15 hold N=l15, M=vgpr; lanes 16-31 hold N=l15, M=vgpr+8.
    // Useful when group(M)==group(N): lanes 0-7 (groups 0,1), lanes 24-31 (groups 2,3).
    const bool useful = (half == 0) ? (l15 < 8) : (l15 >= 8);
    if (useful) {
        const int j   = l15 & 3;      // q component (column of 4x4 moment)
        const bool hi = (grp & 1);    // accumulator VGPR sub-block selector
        float e0 = hi ? acc[4] : acc[0];
        float e1 = hi ? acc[5] : acc[1];
        float e2 = hi ? acc[6] : acc[2];
        float e3 = hi ? acc[7] : acc[3];
        atomicAdd(&blockAcc[0 * 4 + j], e0);
        atomicAdd(&blockAcc[1 * 4 + j], e1);
        atomicAdd(&blockAcc[2 * 4 + j], e2);
        atomicAdd(&blockAcc[3 * 4 + j], e3);
    }
    __syncthreads();
    if (tid < 16) atomicAdd(&mom[b * 16 + tid], blockAcc[tid]);
}

// ---------------------------------------------------------------------------
// Kernel 2: per-batch finalize. Moments -> cov -> 3x3 Jacobi SVD -> rot, t.
// ---------------------------------------------------------------------------
__global__ void finalize_kernel(const float* __restrict__ mom,
                                float* __restrict__ out, int B) {
    const int b = blockIdx.x * blockDim.x + threadIdx.x;
    if (b >= B) return;
    const float* M = mom + b * 16;

    const float W   = M[15];
    const float inv = 1.0f / (W + 1e-6f);
    const float r   = W * inv;

    float cs[3], ct[3], cov[3][3];
    for (int d = 0; d < 3; ++d) { cs[d] = M[d * 4 + 3] * inv; ct[d] = M[12 + d] * inv; }
    for (int d = 0; d < 3; ++d)
        for (int e = 0; e < 3; ++e)
            cov[d][e] = M[d * 4 + e] * inv - cs[d] * ct[e] * (2.0f - r);

    // H = cov^T cov (symmetric), Jacobi eigendecomposition -> V, eigenvalues.
    float H[3][3];
    for (int i = 0; i < 3; ++i)
        for (int j = 0; j < 3; ++j) {
            float s = 0.0f;
            for (int k = 0; k < 3; ++k) s += cov[k][i] * cov[k][j];
            H[i][j] = s;
        }
    float V[3][3] = {{1, 0, 0}, {0, 1, 0}, {0, 0, 1}};
    const int PP[3] = {0, 0, 1}, QQ[3] = {1, 2, 2};
    for (int sweep = 0; sweep < 10; ++sweep) {
        for (int pair = 0; pair < 3; ++pair) {
            const int p = PP[pair], q = QQ[pair];
            const float apq = H[p][q];
            if (fabsf(apq) < 1e-30f) continue;
            const float tau = (H[q][q] - H[p][p]) / (2.0f * apq);
            const float t   = copysignf(1.0f, tau) / (fabsf(tau) + sqrtf(1.0f + tau * tau));
            const float c   = 1.0f / sqrtf(1.0f + t * t);
            const float s   = t * c;
            for (int k = 0; k < 3; ++k) { float a1 = H[k][p], a2 = H[k][q]; H[k][p] = c * a1 - s * a2; H[k][q] = s * a1 + c * a2; }
            for (int k = 0; k < 3; ++k) { float a1 = H[p][k], a2 = H[q][k]; H[p][k] = c * a1 - s * a2; H[q][k] = s * a1 + c * a2; }
            for (int k = 0; k < 3; ++k) { float a1 = V[k][p], a2 = V[k][q]; V[k][p] = c * a1 - s * a2; V[k][q] = s * a1 + c * a2; }
        }
    }
    float ev[3] = {H[0][0], H[1][1], H[2][2]};
    for (int a = 0; a < 2; ++a)
        for (int c = 0; c < 2 - a; ++c)
            if (ev[c] < ev[c + 1]) {
                float tmp = ev[c]; ev[c] = ev[c + 1]; ev[c + 1] = tmp;
                for (int k = 0; k < 3; ++k) { float tv = V[k][c]; V[k][c] = V[k][c + 1]; V[k][c + 1] = tv; }
            }

    // U columns: u_i = cov * v_i / s_i
    float U[3][3];
    for (int i = 0; i < 3; ++i) {
        const float sv   = sqrtf(fmaxf(ev[i], 0.0f));
        const float invs = (sv > 1e-12f) ? (1.0f / sv) : 0.0f;
        for (int k = 0; k < 3; ++k) {
            const float u = cov[k][0] * V[0][i] + cov[k][1] * V[1][i] + cov[k][2] * V[2][i];
            U[k][i] = u * invs;
        }
    }

    // rot = V U^T, with last V column flipped if det <= 0
    float rot[3][3];
    for (int d = 0; d < 3; ++d)
        for (int e = 0; e < 3; ++e)
            rot[d][e] = V[d][0] * U[e][0] + V[d][1] * U[e][1] + V[d][2] * U[e][2];
    const float det = rot[0][0] * (rot[1][1] * rot[2][2] - rot[1][2] * rot[2][1])
                    - rot[0][1] * (rot[1][0] * rot[2][2] - rot[1][2] * rot[2][0])
                    + rot[0][2] * (rot[1][0] * rot[2][1] - rot[1][1] * rot[2][0]);
    if (!(det > 0.0f)) {
        for (int d = 0; d < 3; ++d)
            for (int e = 0; e < 3; ++e)
                rot[d][e] = V[d][0] * U[e][0] + V[d][1] * U[e][1] - V[d][2] * U[e][2];
    }

    for (int d = 0; d < 3; ++d)
        for (int e = 0; e < 3; ++e)
            out[b * 9 + d * 3 + e] = rot[d][e];
    for (int d = 0; d < 3; ++d) {
        const float t = -(rot[d][0] * cs[0] + rot[d][1] * cs[1] + rot[d][2] * cs[2]) + ct[d];
        out[B * 9 + b * 3 + d] = t;
    }
}

// ---------------------------------------------------------------------------
extern "C" void kernel_launch(void* const* d_in, const int* in_sizes, int n_in,
                              void* d_out, int out_size, void* d_ws, size_t ws_size,
                              hipStream_t stream) {
    const float* xyz_s      = (const float*)d_in[0];
    const float* weighted_t = (const float*)d_in[1];
    const float* weights    = (const float*)d_in[2];
    const float* label      = (const float*)d_in[3];
    float* out = (float*)d_out;
    float* mom = (float*)d_ws;

    const int B = out_size / 12;            // rot(9) + translation(3) per batch
    const int N = in_sizes[2] / B;          // weights is (B,1,N)

    // 1) zero moment accumulators (workspace is poisoned, must init every call)
    const int nacc = B * 16;
    zero_ws_kernel<<<(nacc + 255) / 256, 256, 0, stream>>>(mom, nacc);

    // 2) streaming moment reduction (bandwidth-bound; WMMA does the 4x4 GEMM)
    dim3 grid(32, B);
    moments_kernel<<<grid, WAVES * 32, 0, stream>>>(xyz_s, weighted_t, weights,
                                                    label, mom, N);

    // 3) tiny SVD + pose epilogue
    finalize_kernel<<<(B + 31) / 32, 32, 0, stream>>>(mom, out, B);
}